// S2VTModel_89361089561134
// MI455X (gfx1250) — compile-verified
//
#include <hip/hip_runtime.h>
#include <stdint.h>

// Problem constants (from reference)
#define BATCH   128
#define T_ENC   40
#define T_DEC   27          // target[:, :-1] -> 27 decode steps
#define HID     512
#define EMB     1024
#define VOC     20000

typedef __attribute__((ext_vector_type(2))) float v2f;  // A/B of 16x16x4 f32 WMMA
typedef __attribute__((ext_vector_type(8))) float v8f;  // C/D accumulator
typedef __attribute__((ext_vector_type(4))) unsigned int u32x4;
typedef __attribute__((ext_vector_type(4))) int i32x4;
typedef __attribute__((ext_vector_type(8))) int i32x8;

#if defined(__has_builtin)
#if __has_builtin(__builtin_amdgcn_tensor_load_to_lds)
#define HAVE_TDM 1
#endif
#endif
#ifndef HAVE_TDM
#define HAVE_TDM 0
#endif

// ---------------------------------------------------------------------------
// WMMA helpers: D = A(16x4) x B(4x16) + C, fp32 exact (V_WMMA_F32_16X16X4_F32)
// A layout: lane l, reg v  -> element (m = l&15, k = 2*(l>>4) + v)
// B layout: lane l, reg v  -> element (k = 2*(l>>4) + v, n = l&15)  [B[k][n] = W[n][k]]
// C layout: lane l, reg v  -> element (m = v + 8*(l>>4), n = l&15)
// ---------------------------------------------------------------------------
__device__ __forceinline__ v8f wmma_f32(v2f a, v2f b, v8f c) {
    return __builtin_amdgcn_wmma_f32_16x16x4_f32(
        /*neg_a=*/false, a, /*neg_b=*/false, b,
        /*c_mod=*/(short)0, c, /*reuse_a=*/false, /*reuse_b=*/false);
}

__device__ __forceinline__ v2f load_a(const float* __restrict__ X, int lda,
                                      int m0, int k0, int lane) {
    int row = m0 + (lane & 15);
    int kh  = (lane >> 4) << 1;            // 0 or 2
    const float* p = X + (size_t)row * lda + k0 + kh;
    v2f a; a.x = p[0]; a.y = p[1];
    return a;
}

// B tile comes from row-major weight W (rows = output cols n, cols = k):
// B[k][n] = W[n][k] -> column n of B is a contiguous row of W.
__device__ __forceinline__ v2f load_b(const float* __restrict__ W, int ldw,
                                      int n0, int k0, int lane) {
    int col = n0 + (lane & 15);
    int kh  = (lane >> 4) << 1;
    const float* p = W + (size_t)col * ldw + k0 + kh;
    v2f b; b.x = p[0]; b.y = p[1];
    return b;
}

// Accumulate A(rows m0..m0+15 of src, K wide) x W^T into the 3 gate tiles
// (r, z, n) at output columns {n0, n0+512, n0+1024} of the 3H-wide gate matrix.
__device__ __forceinline__ void gate_gemm(const float* __restrict__ A, int lda,
                                          int m0, int K,
                                          const float* __restrict__ W, int ldw,
                                          int wkoff, int n0,
                                          v8f& ar, v8f& az, v8f& an, int lane) {
    for (int k = 0; k < K; k += 4) {
        v2f a  = load_a(A, lda, m0, k, lane);
        v2f br = load_b(W, ldw, 0 * HID + n0, wkoff + k, lane);
        v2f bz = load_b(W, ldw, 1 * HID + n0, wkoff + k, lane);
        v2f bn = load_b(W, ldw, 2 * HID + n0, wkoff + k, lane);
        ar = wmma_f32(a, br, ar);
        az = wmma_f32(a, bz, az);
        an = wmma_f32(a, bn, an);
    }
}

// ---------------------------------------------------------------------------
// One fused GRU step for one layer:
//   gates = X0 @ Wih[:, :K0]^T + X1 @ Wih[:, x1off:x1off+K1]^T + H @ Whh^T
//   h_new = GRU update (r/z/n)
// Block = 4 waves, each wave computes a 16x16 tile of h_new (block tile 16x64).
// ---------------------------------------------------------------------------
__global__ void __launch_bounds__(128)
gru_step_kernel(const float* __restrict__ X0, int ld0, int K0,
                const float* __restrict__ X1, int K1, int x1off,
                const float* __restrict__ Hin,
                const float* __restrict__ Wih, int ldwih,
                const float* __restrict__ Whh,
                const float* __restrict__ bih,
                const float* __restrict__ bhh,
                float* __restrict__ Hout,
                float* __restrict__ Hseq) {
    const int lane = threadIdx.x & 31;
    const int wave = threadIdx.x >> 5;
    const int m0   = blockIdx.y * 16;
    const int n0   = blockIdx.x * 64 + wave * 16;

    v8f ar = {}, az = {}, ain = {}, ahn = {};

    if (K0 > 0) gate_gemm(X0, ld0, m0, K0, Wih, ldwih, 0,     n0, ar, az, ain, lane);
    if (K1 > 0) gate_gemm(X1, K1,  m0, K1, Wih, ldwih, x1off, n0, ar, az, ain, lane);
    gate_gemm(Hin, HID, m0, HID, Whh, HID, 0, n0, ar, az, ahn, lane);

    const int nc    = n0 + (lane & 15);
    const int mbase = m0 + ((lane >> 4) << 3);
    const float br_  = bih[nc]           + bhh[nc];
    const float bz_  = bih[nc + HID]     + bhh[nc + HID];
    const float bin_ = bih[nc + 2 * HID];
    const float bhn_ = bhh[nc + 2 * HID];
#pragma unroll
    for (int v = 0; v < 8; ++v) {
        const int m = mbase + v;
        float r  = 1.f / (1.f + __expf(-(ar[v] + br_)));
        float z  = 1.f / (1.f + __expf(-(az[v] + bz_)));
        float nn = tanhf(ain[v] + bin_ + r * (ahn[v] + bhn_));
        float hp = Hin[(size_t)m * HID + nc];
        float hnew = (1.f - z) * nn + z * hp;
        Hout[(size_t)m * HID + nc] = hnew;
        if (Hseq) Hseq[(size_t)m * HID + nc] = hnew;
    }
}

// ---------------------------------------------------------------------------
// Gather decoder word embeddings: words[t][b][:] = E[target[b][t]][:]
// ---------------------------------------------------------------------------
__global__ void gather_words_kernel(const float* __restrict__ E,
                                    const long long* __restrict__ tgt,
                                    float* __restrict__ words) {
    const int t = blockIdx.x;   // 0..26
    const int b = blockIdx.y;   // 0..127
    long long idx = tgt[(size_t)b * (T_DEC + 1) + t];
    if (idx < 0) idx = 0;
    if (idx >= VOC) idx = VOC - 1;
    const float* src = E + (size_t)idx * HID;
    float* dst = words + ((size_t)t * BATCH + b) * HID;
    for (int i = threadIdx.x; i < HID; i += blockDim.x) dst[i] = src[i];
}

// ---------------------------------------------------------------------------
// Projection GEMM with TDM double-buffered LDS staging.
// C(3456 x 20000) = h2_seq (3456 x 512) @ W_out^T + b_out
// Block = 8 waves (256 thr), block tile 64(M) x 128(N); wave tile 32x32.
// K chunks of 32; A tile 64x32 and B tile 128x32 DMA'd to LDS by the
// Tensor Data Mover with pad_interval=32dw / pad_amount=2dw, giving a
// conflict-free 34-float LDS row stride.
// ---------------------------------------------------------------------------
#define PJ_BM 64
#define PJ_BN 128
#define PJ_KC 32
#define LDS_STRIDE 34                          // 32 data dwords + 2 pad dwords
#define A_TILE_FLOATS (PJ_BM * LDS_STRIDE)     // 2176
#define B_TILE_FLOATS (PJ_BN * LDS_STRIDE)     // 4352
#define PJ_SHMEM_BYTES ((2 * A_TILE_FLOATS + 2 * B_TILE_FLOATS) * 4)   // 52224

#if HAVE_TDM
// Issue one 2D TDM tile load: tile_rows x tile_k elements (f32) from a
// row-major tensor (row stride = row_stride elems) into LDS at lds_byte_off,
// with a 2-dword pad inserted after every 32 dwords (-> 34-float LDS rows).
// rows_rem/k_rem = remaining tensor extents from the tile start (HW clamps
// OOB reads to zero).  D# packing per CDNA5 ISA Sec 8.3/8.4.
__device__ __forceinline__ void tdm_load_2d(unsigned lds_byte_off,
                                            const float* gptr,
                                            unsigned tile_k, unsigned tile_rows,
                                            unsigned rows_rem, unsigned k_rem,
                                            unsigned row_stride) {
    union { unsigned int w[4]; u32x4 v; } g0;
    unsigned long long ga = (unsigned long long)(uintptr_t)gptr;
    g0.w[0] = 1u;                                              // count=1 (valid D#)
    g0.w[1] = lds_byte_off;                                    // lds_addr
    g0.w[2] = (unsigned)ga;                                    // global_addr lo
    g0.w[3] = (unsigned)((ga >> 32) & 0x01FFFFFFu) | (2u << 30); // addr hi | type=2

    unsigned long long q0 = 0, q1 = 0, q2 = 0, q3 = 0;
    // data_size=2 (4B) | pad_enable | pad_interval=4 (32 dw) | pad_amount=1 (2 dw)
    q0 |= (2ull << 16) | (1ull << 20) | (4ull << 22) | (1ull << 25);
    q0 |= ((unsigned long long)(k_rem & 0xFFFFu)) << 48;       // tensor_dim0[15:0]
    q1 |= (unsigned long long)(k_rem >> 16);                   // tensor_dim0[31:16]
    q1 |= ((unsigned long long)rows_rem) << 16;                // tensor_dim1
    q1 |= ((unsigned long long)tile_k) << 48;                  // tile_dim0
    q2 |= (unsigned long long)tile_rows;                       // tile_dim1 (tile_dim2=0)
    q2 |= ((unsigned long long)row_stride) << 32;              // tensor_dim0_stride lo32
    // q3: stride hi16 = 0, tensor_dim1_stride = 0 (2D)
    union { unsigned long long q[4]; i32x8 v; } g1;
    g1.q[0] = q0; g1.q[1] = q1; g1.q[2] = q2; g1.q[3] = q3;
    i32x4 z4 = {0, 0, 0, 0};
    i32x8 z8 = {0, 0, 0, 0, 0, 0, 0, 0};
    // 6-arg form (amdgpu-toolchain / clang-23 signature)
    __builtin_amdgcn_tensor_load_to_lds(g0.v, g1.v, z4, z4, z8, 0);
}
#endif

__device__ __forceinline__ v2f lds_tile(const float* base, int row0, int k, int lane) {
    int row = row0 + (lane & 15);
    int kk  = k + ((lane >> 4) << 1);
    const float* p = base + row * LDS_STRIDE + kk;
    v2f r; r.x = p[0]; r.y = p[1];
    return r;
}

__global__ void __launch_bounds__(256)
proj_kernel(const float* __restrict__ A,
            const float* __restrict__ Wout,
            const float* __restrict__ bout,
            float* __restrict__ out) {
    extern __shared__ float smem[];
    const int lane = threadIdx.x & 31;
    const int wave = threadIdx.x >> 5;
    const int m0 = blockIdx.y * PJ_BM;
    const int n0 = blockIdx.x * PJ_BN;
    const int mw = (wave >> 2) * 32;            // 0 / 32
    const int nw = (wave & 3) * 32;             // 0 / 32 / 64 / 96

    const unsigned aoff[2] = { 0u, (unsigned)A_TILE_FLOATS };
    const unsigned boff[2] = { (unsigned)(2 * A_TILE_FLOATS),
                               (unsigned)(2 * A_TILE_FLOATS + B_TILE_FLOATS) };
    const unsigned rows_remA = (unsigned)(T_DEC * BATCH - m0);
    const unsigned rows_remB = (unsigned)(VOC - n0);
    const int NCHUNK = HID / PJ_KC;             // 16

#if HAVE_TDM
    if (wave == 0) {
        tdm_load_2d(aoff[0] * 4u, A    + (size_t)m0 * HID, PJ_KC, PJ_BM, rows_remA, HID, HID);
        tdm_load_2d(boff[0] * 4u, Wout + (size_t)n0 * HID, PJ_KC, PJ_BN, rows_remB, HID, HID);
    }
#endif

    v8f acc00 = {}, acc01 = {}, acc10 = {}, acc11 = {};

    for (int c = 0; c < NCHUNK; ++c) {
#if HAVE_TDM
        if (wave == 0) __builtin_amdgcn_s_wait_tensorcnt(0);
        __syncthreads();
        if (wave == 0 && c + 1 < NCHUNK) {
            const int kn = (c + 1) * PJ_KC;
            tdm_load_2d(aoff[(c + 1) & 1] * 4u, A    + (size_t)m0 * HID + kn,
                        PJ_KC, PJ_BM, rows_remA, HID - kn, HID);
            tdm_load_2d(boff[(c + 1) & 1] * 4u, Wout + (size_t)n0 * HID + kn,
                        PJ_KC, PJ_BN, rows_remB, HID - kn, HID);
        }
#else
        // Fallback: cooperative global->LDS staging (no TDM builtin available)
        {
            const int k0 = c * PJ_KC;
            for (int i = threadIdx.x; i < PJ_BM * PJ_KC / 2; i += 256) {
                int r = i / (PJ_KC / 2), kk = (i % (PJ_KC / 2)) * 2;
                int gr = m0 + r;
                smem[aoff[c & 1] + r * LDS_STRIDE + kk]     = A[(size_t)gr * HID + k0 + kk];
                smem[aoff[c & 1] + r * LDS_STRIDE + kk + 1] = A[(size_t)gr * HID + k0 + kk + 1];
            }
            for (int i = threadIdx.x; i < PJ_BN * PJ_KC / 2; i += 256) {
                int r = i / (PJ_KC / 2), kk = (i % (PJ_KC / 2)) * 2;
                int gr = n0 + r; if (gr >= VOC) gr = VOC - 1;
                smem[boff[c & 1] + r * LDS_STRIDE + kk]     = Wout[(size_t)gr * HID + k0 + kk];
                smem[boff[c & 1] + r * LDS_STRIDE + kk + 1] = Wout[(size_t)gr * HID + k0 + kk + 1];
            }
            __syncthreads();
        }
#endif
        const float* Ab = smem + aoff[c & 1];
        const float* Bb = smem + boff[c & 1];
#pragma unroll
        for (int kk = 0; kk < PJ_KC; kk += 4) {
            v2f a0 = lds_tile(Ab, mw,      kk, lane);
            v2f a1 = lds_tile(Ab, mw + 16, kk, lane);
            v2f b0 = lds_tile(Bb, nw,      kk, lane);
            v2f b1 = lds_tile(Bb, nw + 16, kk, lane);
            acc00 = wmma_f32(a0, b0, acc00);
            acc01 = wmma_f32(a0, b1, acc01);
            acc10 = wmma_f32(a1, b0, acc10);
            acc11 = wmma_f32(a1, b1, acc11);
        }
        __syncthreads();
    }

    // Epilogue: bias + scatter to out[b][t][v]
#pragma unroll
    for (int sm = 0; sm < 2; ++sm) {
#pragma unroll
        for (int sn = 0; sn < 2; ++sn) {
            v8f& acc = sm == 0 ? (sn == 0 ? acc00 : acc01)
                               : (sn == 0 ? acc10 : acc11);
            const int nc = n0 + nw + sn * 16 + (lane & 15);
            if (nc < VOC) {
                const float bb = bout[nc];
                const int mbase = m0 + mw + sm * 16 + ((lane >> 4) << 3);
#pragma unroll
                for (int v = 0; v < 8; ++v) {
                    const int m  = mbase + v;
                    const int t  = m >> 7;
                    const int bi = m & 127;
                    out[((size_t)bi * T_DEC + t) * VOC + nc] = acc[v] + bb;
                }
            }
        }
    }
}

// ---------------------------------------------------------------------------
// In-place row-wise log-softmax over the vocab dim. One block per (t, b) row.
// ---------------------------------------------------------------------------
__global__ void __launch_bounds__(256)
logsoftmax_kernel(float* __restrict__ out) {
    __shared__ float red[256];
    const int r = blockIdx.x;           // r = t*128 + b
    const int t = r >> 7;
    const int b = r & 127;
    float* row = out + ((size_t)b * T_DEC + t) * VOC;

    float mx = -3.0e38f;
    for (int i = threadIdx.x; i < VOC; i += 256) mx = fmaxf(mx, row[i]);
    red[threadIdx.x] = mx;
    __syncthreads();
    for (int s = 128; s > 0; s >>= 1) {
        if (threadIdx.x < s) red[threadIdx.x] = fmaxf(red[threadIdx.x], red[threadIdx.x + s]);
        __syncthreads();
    }
    mx = red[0];
    __syncthreads();

    float sum = 0.f;
    for (int i = threadIdx.x; i < VOC; i += 256) sum += __expf(row[i] - mx);
    red[threadIdx.x] = sum;
    __syncthreads();
    for (int s = 128; s > 0; s >>= 1) {
        if (threadIdx.x < s) red[threadIdx.x] += red[threadIdx.x + s];
        __syncthreads();
    }
    const float lse = mx + __logf(red[0]);
    __syncthreads();

    for (int i = threadIdx.x; i < VOC; i += 256) row[i] = row[i] - lse;
}

__global__ void zero_kernel(float* __restrict__ p, int n) {
    int i = blockIdx.x * 256 + threadIdx.x;
    if (i < n) p[i] = 0.f;
}

// ---------------------------------------------------------------------------
extern "C" void kernel_launch(void* const* d_in, const int* in_sizes, int n_in,
                              void* d_out, int out_size, void* d_ws, size_t ws_size,
                              hipStream_t stream) {
    (void)in_sizes; (void)n_in; (void)out_size; (void)ws_size;
    const float*     input  = (const float*)d_in[0];
    const long long* target = (const long long*)d_in[1];   // jnp.int64 per reference
    const float* W_ih1 = (const float*)d_in[2];
    const float* W_hh1 = (const float*)d_in[3];
    const float* b_ih1 = (const float*)d_in[4];
    const float* b_hh1 = (const float*)d_in[5];
    const float* W_ih2 = (const float*)d_in[6];
    const float* W_hh2 = (const float*)d_in[7];
    const float* b_ih2 = (const float*)d_in[8];
    const float* b_hh2 = (const float*)d_in[9];
    const float* E     = (const float*)d_in[10];
    const float* W_out = (const float*)d_in[11];
    const float* b_out = (const float*)d_in[12];
    float* out = (float*)d_out;

    // Workspace layout (floats): 2x h1 + 2x h2 ping-pong, words, h2_seq (~15.2 MB)
    float* ws = (float*)d_ws;
    float* h1[2] = { ws,                   ws + 1 * BATCH * HID };
    float* h2[2] = { ws + 2 * BATCH * HID, ws + 3 * BATCH * HID };
    float* words = ws + 4 * (size_t)BATCH * HID;
    float* h2seq = words + (size_t)T_DEC * BATCH * HID;

    {
        const int n = 4 * BATCH * HID;
        zero_kernel<<<(n + 255) / 256, 256, 0, stream>>>(ws, n);
    }

    const dim3 grid_g(HID / 64, BATCH / 16);   // 8 x 8 blocks, 4 waves each

    int cur = 0;
    // ---- Encoder: 40 steps ----
    for (int t = 0; t < T_ENC; ++t) {
        const int nxt = cur ^ 1;
        gru_step_kernel<<<grid_g, 128, 0, stream>>>(
            input + (size_t)t * EMB, T_ENC * EMB, EMB,
            nullptr, 0, 0,
            h1[cur], W_ih1, EMB, W_hh1, b_ih1, b_hh1,
            h1[nxt], nullptr);
        // Layer 2: concat(h1, 0) -> only first 512 columns of W_ih2 matter
        gru_step_kernel<<<grid_g, 128, 0, stream>>>(
            h1[nxt], HID, HID,
            nullptr, 0, 0,
            h2[cur], W_ih2, EMB, W_hh2, b_ih2, b_hh2,
            h2[nxt], nullptr);
        cur = nxt;
    }

    gather_words_kernel<<<dim3(T_DEC, BATCH), 128, 0, stream>>>(E, target, words);

    // ---- Decoder: 27 steps ----
    for (int t = 0; t < T_DEC; ++t) {
        const int nxt = cur ^ 1;
        // Layer 1: zero frame input -> gi = b_ih1 only (skip X GEMM)
        gru_step_kernel<<<grid_g, 128, 0, stream>>>(
            nullptr, 0, 0,
            nullptr, 0, 0,
            h1[cur], W_ih1, EMB, W_hh1, b_ih1, b_hh1,
            h1[nxt], nullptr);
        // Layer 2: concat(h1, word_t) -> both halves of W_ih2
        gru_step_kernel<<<grid_g, 128, 0, stream>>>(
            h1[nxt], HID, HID,
            words + (size_t)t * BATCH * HID, HID, HID,
            h2[cur], W_ih2, EMB, W_hh2, b_ih2, b_hh2,
            h2[nxt], h2seq + (size_t)t * BATCH * HID);
        cur = nxt;
    }

    // ---- Batched projection with TDM-staged tiles ----
    proj_kernel<<<dim3((VOC + PJ_BN - 1) / PJ_BN, (T_DEC * BATCH) / PJ_BM),
                  256, PJ_SHMEM_BYTES, stream>>>(h2seq, W_out, b_out, out);

    // ---- log_softmax in place over vocab dim ----
    logsoftmax_kernel<<<T_DEC * BATCH, 256, 0, stream>>>(out);
}